// CustomMatrixMultiplication_13511967113734
// MI455X (gfx1250) — compile-verified
//
#include <hip/hip_runtime.h>

// Problem constants (fixed by the reference): B=64, N=M=P=1024.
#define BB 64
#define NN 1024
#define MM 1024
#define PP 1024

typedef __attribute__((ext_vector_type(2))) float v2f;
typedef __attribute__((ext_vector_type(8))) float v8f;

// ---------------------------------------------------------------------------
// Stage 1: s[b, n] = sum_p m2[b, p, n]
// One wave handles a 16-wide column tile (b, n0..n0+15) and sweeps all P rows.
// WMMA form: D += ones(16x4) x B(4x16), so every D row equals the column sums.
// Slot map (both A and B): (vgpr j, lane-half h) -> K = 2h + j.
// ---------------------------------------------------------------------------
__global__ __launch_bounds__(256) void sumP_wmma(const float* __restrict__ m2,
                                                 float* __restrict__ s) {
  const int wave = threadIdx.x >> 5;
  const int lane = threadIdx.x & 31;
  const int tile = blockIdx.x * 8 + wave;        // 4096 tiles = B * N/16
  const int b    = tile >> 6;                    // tile / (N/16)
  const int n0   = (tile & 63) << 4;             // 16-col tile base
  const int half = lane >> 4;                    // 0: lanes 0-15, 1: lanes 16-31
  const int col  = lane & 15;

  const float* __restrict__ base =
      m2 + (size_t)b * (size_t)(PP * NN) + (size_t)(n0 + col);

  v2f ones;
  ones.x = 1.0f;
  ones.y = 1.0f;

  v8f acc = {};
  for (int p = 0; p < PP; p += 16) {
#pragma unroll
    for (int q = 0; q < 4; ++q) {
      const int r = p + q * 4 + 2 * half;        // rows for K = 2h (+1 for K=2h+1)
      v2f bm;
      bm.x = base[(size_t)r * NN];
      bm.y = base[(size_t)(r + 1) * NN];
      acc = __builtin_amdgcn_wmma_f32_16x16x4_f32(
          false, ones, false, bm, (short)0, acc, false, false);
    }
  }

  // D VGPR0, lanes 0-15 = (M=0, N=lane): the column sums.
  if (lane < 16) s[(size_t)b * NN + (size_t)(n0 + lane)] = acc[0];
}

// ---------------------------------------------------------------------------
// Stage 2: out[b, m] = sum_n m1[b, n, m] * s[b, n]
// One wave handles a 16-wide m tile and sweeps all N rows. A carries the four
// s-values of the current K slice, replicated down all 16 rows; B is the
// matching 4x16 tile of m1. Row 0 of D is the answer.
// ---------------------------------------------------------------------------
__global__ __launch_bounds__(256) void matvec_wmma(const float* __restrict__ m1,
                                                   const float* __restrict__ s,
                                                   float* __restrict__ out) {
  const int wave = threadIdx.x >> 5;
  const int lane = threadIdx.x & 31;
  const int tile = blockIdx.x * 8 + wave;        // 4096 tiles = B * M/16
  const int b    = tile >> 6;
  const int m0   = (tile & 63) << 4;
  const int half = lane >> 4;
  const int col  = lane & 15;

  const float* __restrict__ m1b =
      m1 + (size_t)b * (size_t)(NN * MM) + (size_t)(m0 + col);
  const float* __restrict__ sb = s + (size_t)b * NN;

  v8f acc = {};
  for (int n = 0; n < NN; n += 16) {
#pragma unroll
    for (int q = 0; q < 4; ++q) {
      const int r = n + q * 4 + 2 * half;        // K = 2h row; +1 => K = 2h+1
      v2f a, bm;
      a.x  = sb[r];                              // s value paired with bm.x row
      a.y  = sb[r + 1];                          // s value paired with bm.y row
      bm.x = m1b[(size_t)r * MM];
      bm.y = m1b[(size_t)(r + 1) * MM];
      acc = __builtin_amdgcn_wmma_f32_16x16x4_f32(
          false, a, false, bm, (short)0, acc, false, false);
    }
  }

  if (lane < 16) out[(size_t)b * MM + (size_t)(m0 + lane)] = acc[0];
}

extern "C" void kernel_launch(void* const* d_in, const int* in_sizes, int n_in,
                              void* d_out, int out_size, void* d_ws, size_t ws_size,
                              hipStream_t stream) {
  const float* m1 = (const float*)d_in[0];   // [B, N, M]
  const float* m2 = (const float*)d_in[1];   // [B, P, N]
  float* out = (float*)d_out;                // [B, M]
  float* s   = (float*)d_ws;                 // [B, N] scratch (256 KB)

  // 4096 waves per stage, 8 waves (256 threads) per block -> 512 blocks.
  sumP_wmma<<<512, 256, 0, stream>>>(m2, s);
  matvec_wmma<<<512, 256, 0, stream>>>(m1, s, out);
}